// ParallelDLIEMA_17789754541002
// MI455X (gfx1250) — compile-verified
//
#include <hip/hip_runtime.h>

typedef __attribute__((ext_vector_type(2))) float v2f;
typedef __attribute__((ext_vector_type(8))) float v8f;

#define T_DIM 4096
#define B_DIM 8
#define E_DIM 512
#define C16   (T_DIM / 16)   // 256 chunks of 16 timesteps
#define ETILE 32             // channels per workgroup in K3
#define SUPER 256            // timesteps per workgroup in K3 (16 sub-chunks)

// ---- CDNA5 async global<->LDS path (guarded; falls back to plain ld/st) ---
#if defined(__gfx1250__) && __has_builtin(__builtin_amdgcn_global_load_async_to_lds_b32)
#define USE_ASYNC_G2L 1
#else
#define USE_ASYNC_G2L 0
#endif
#if defined(__gfx1250__) && __has_builtin(__builtin_amdgcn_global_store_async_from_lds_b32)
#define USE_ASYNC_L2G 1
#else
#define USE_ASYNC_L2G 0
#endif

#if USE_ASYNC_G2L
__device__ __forceinline__ void async_g2l_b32(const float* g, float* l) {
    __builtin_amdgcn_global_load_async_to_lds_b32(
        (__attribute__((address_space(1))) int*)(g),
        (__attribute__((address_space(3))) int*)(l), 0, 0);
}
#endif
#if USE_ASYNC_L2G
__device__ __forceinline__ void async_l2g_b32(float* g, const float* l) {
    __builtin_amdgcn_global_store_async_from_lds_b32(
        (__attribute__((address_space(1))) int*)(g),
        (__attribute__((address_space(3))) int*)(l), 0, 0);
}
#endif
__device__ __forceinline__ void wait_async0() {
#if defined(__gfx1250__) && (USE_ASYNC_G2L || USE_ASYNC_L2G)
#if __has_builtin(__builtin_amdgcn_s_wait_asynccnt)
    __builtin_amdgcn_s_wait_asynccnt(0);
#else
    asm volatile("s_wait_asynccnt 0" ::: "memory");
#endif
#endif
}

__device__ __forceinline__ float sigmoidf_(float v) { return 1.f / (1.f + __expf(-v)); }

// beta^(d&31) via repeated-squaring selects — branch-free, no dynamic indexing
__device__ __forceinline__ float ipow_bits(float b1, float b2, float b4, float b8,
                                           float b16, int d) {
    float p = ((d & 1) ? b1 : 1.f) * ((d & 2) ? b2 : 1.f);
    p *= ((d & 4) ? b4 : 1.f) * ((d & 8) ? b8 : 1.f);
    p *= ((d & 16) ? b16 : 1.f);
    return p;
}
// beta^d for d possibly negative -> 0 (single final cndmask; bases in (0,1) so
// the discarded clamped-exponent product is always finite)
__device__ __forceinline__ float tripow(float b1, float b2, float b4, float b8,
                                        float b16, int d) {
    float p = ipow_bits(b1, b2, b4, b8, b16, d & 31);
    return (d < 0) ? 0.f : p;
}

// ---------------- K1: per-16-chunk local-scan summaries --------------------
__global__ void k1_summaries(const float* __restrict__ x,
                             const float* __restrict__ rawp,
                             const float* __restrict__ rawn,
                             float2* __restrict__ S) {   // [c][b][e]
    int idx = blockIdx.x * blockDim.x + threadIdx.x;  // c*B*E + b*E + e
    int e  = idx % E_DIM;
    int be = idx / E_DIM;
    int b  = be % B_DIM;
    int c  = be / B_DIM;
    float bp = sigmoidf_(rawp[e]);
    float bn = sigmoidf_(rawn[e]);
    float gp = 1.f - bp, gn = 1.f - bn;
    const float* px = x + ((size_t)(c * 16) * B_DIM + b) * E_DIM + e;
    float sp = 0.f, sn = 0.f;
#pragma unroll
    for (int k = 0; k < 16; ++k) {
        float xv = px[(size_t)k * B_DIM * E_DIM];
        sp = bp * sp + fmaxf(xv, 0.f) * gp;
        sn = bn * sn + fminf(xv, 0.f) * gn;
    }
    S[idx] = make_float2(sp, sn);
}

// ---------------- K2: sequential chunk-carry combine per channel -----------
__global__ void k2_carries(const float* __restrict__ rawp,
                           const float* __restrict__ rawn,
                           const float2* __restrict__ S,      // [c][b][e]
                           float2* __restrict__ carry) {      // [b][e][c]
    int idx = blockIdx.x * blockDim.x + threadIdx.x;  // b*E + e
    if (idx >= B_DIM * E_DIM) return;
    int e = idx % E_DIM;
    int b = idx / E_DIM;
    float bp = sigmoidf_(rawp[e]);
    float bn = sigmoidf_(rawn[e]);
    float bp16 = bp, bn16 = bn;
#pragma unroll
    for (int i = 0; i < 4; ++i) { bp16 *= bp16; bn16 *= bn16; }
    float mp = 0.f, mn = 0.f;
    float2* cdst = carry + ((size_t)b * E_DIM + e) * C16;
    for (int c = 0; c < C16; ++c) {
        cdst[c] = make_float2(mp, mn);
        float2 s = S[((size_t)c * B_DIM + b) * E_DIM + e];
        mp = bp16 * mp + s.x;
        mn = bn16 * mn + s.y;
    }
}

// ---------------- K3: WMMA local scans + carry correction ------------------
__global__ __launch_bounds__(1024) void k3_wmma_scan(
    const float* __restrict__ x,
    const float* __restrict__ rawp,
    const float* __restrict__ rawn,
    const float2* __restrict__ carry,   // [b][e][c]
    float* __restrict__ out) {
    __shared__ float lds[ETILE * SUPER];  // [w][t], 32 KB

    const int tid = threadIdx.x;
    const int net = E_DIM / ETILE;            // 16 e-tiles
    int et  = blockIdx.x % net;
    int tmp = blockIdx.x / net;
    int b   = tmp % B_DIM;
    int sc  = tmp / B_DIM;                    // superchunk 0..15
    const int e0 = et * ETILE;
    const int t0 = sc * SUPER;

    // --- stage x[t0..t0+255][b][e0..e0+31] -> lds[w][t] (128B-coalesced
    //     global side; transpose happens via per-lane LDS addresses) ---
    {
        const float* src = x + ((size_t)t0 * B_DIM + b) * E_DIM + e0;
#pragma unroll
        for (int it = 0; it < (ETILE * SUPER) / 1024; ++it) {
            int flat = tid + it * 1024;
            int t = flat >> 5;       // / ETILE
            int w = flat & 31;
#if USE_ASYNC_G2L
            async_g2l_b32(&src[(size_t)t * B_DIM * E_DIM + w], &lds[w * SUPER + t]);
#else
            lds[w * SUPER + t] = src[(size_t)t * B_DIM * E_DIM + w];
#endif
        }
#if USE_ASYNC_G2L
        wait_async0();
#endif
    }
    __syncthreads();

    const int w    = tid >> 5;       // wave id = channel offset in tile
    const int lane = tid & 31;
    const int e    = e0 + w;
    const int n    = lane & 15;      // column (sub-chunk) / A-matrix row M
    const int hi   = lane >> 4;      // lane half selects K pair {0,1} vs {2,3}

    float bp1 = sigmoidf_(rawp[e]);
    float bn1 = sigmoidf_(rawn[e]);
    float bp2 = bp1 * bp1, bp4 = bp2 * bp2, bp8 = bp4 * bp4, bp16 = bp8 * bp8;
    float bn2 = bn1 * bn1, bn4 = bn2 * bn2, bn8 = bn4 * bn4, bn16 = bn8 * bn8;
    float gp = 1.f - bp1, gn = 1.f - bn1;

    // per-lane U values: rows kk*4 + 2*hi + {0,1} of column n (t = n*16+row),
    // loaded as aligned float2 (ds_load_b64)
    float xv[8];
#pragma unroll
    for (int kk = 0; kk < 4; ++kk) {
        int row = kk * 4 + 2 * hi;
        float2 xy = *(const float2*)&lds[w * SUPER + n * 16 + row];
        xv[kk * 2 + 0] = xy.x;
        xv[kk * 2 + 1] = xy.y;
    }

    // carry entering sub-chunk c = sc*16 + n (lanes n and n+16 broadcast-load)
    float2 cy = carry[((size_t)b * E_DIM + e) * C16 + sc * 16 + n];

    v8f acc = {0.f, 0.f, 0.f, 0.f, 0.f, 0.f, 0.f, 0.f};
#pragma unroll
    for (int kk = 0; kk < 4; ++kk) {  // positive branch: L_pos @ U_pos
        int Ka = kk * 4 + 2 * hi;
        v2f a, bb;
        a.x = tripow(bp1, bp2, bp4, bp8, bp16, n - Ka);
        a.y = tripow(bp1, bp2, bp4, bp8, bp16, n - Ka - 1);
        bb.x = fmaxf(xv[kk * 2 + 0], 0.f) * gp;
        bb.y = fmaxf(xv[kk * 2 + 1], 0.f) * gp;
        acc = __builtin_amdgcn_wmma_f32_16x16x4_f32(false, a, false, bb,
                                                    (short)0, acc, false, false);
    }
#pragma unroll
    for (int kk = 0; kk < 4; ++kk) {  // negative branch accumulates into same C
        int Ka = kk * 4 + 2 * hi;
        v2f a, bb;
        a.x = tripow(bn1, bn2, bn4, bn8, bn16, n - Ka);
        a.y = tripow(bn1, bn2, bn4, bn8, bn16, n - Ka - 1);
        bb.x = fminf(xv[kk * 2 + 0], 0.f) * gn;
        bb.y = fminf(xv[kk * 2 + 1], 0.f) * gn;
        acc = __builtin_amdgcn_wmma_f32_16x16x4_f32(false, a, false, bb,
                                                    (short)0, acc, false, false);
    }

    // corrections: out[i] = Y[i] + bp^(i+1)*carry_p + bn^(i+1)*carry_n,
    // written back to this wave's LDS row as two aligned float4 (ds_store_b128)
    {
        float o[8];
#pragma unroll
        for (int r = 0; r < 8; ++r) {
            int M = r + 8 * hi;  // C/D layout: VGPR r -> row M (lanes 16-31: +8)
            float corr = ipow_bits(bp1, bp2, bp4, bp8, bp16, M + 1) * cy.x +
                         ipow_bits(bn1, bn2, bn4, bn8, bn16, M + 1) * cy.y;
            o[r] = acc[r] + corr;
        }
        float4* dst128 = (float4*)&lds[w * SUPER + n * 16 + 8 * hi];
        dst128[0] = make_float4(o[0], o[1], o[2], o[3]);
        dst128[1] = make_float4(o[4], o[5], o[6], o[7]);
    }
    __syncthreads();

    // --- drain lds[w][t] -> out[t][b][e] (128B-coalesced global side) ---
    {
        float* dst = out + ((size_t)t0 * B_DIM + b) * E_DIM + e0;
#pragma unroll
        for (int it = 0; it < (ETILE * SUPER) / 1024; ++it) {
            int flat = tid + it * 1024;
            int t = flat >> 5;
            int we = flat & 31;
#if USE_ASYNC_L2G
            async_l2g_b32(&dst[(size_t)t * B_DIM * E_DIM + we], &lds[we * SUPER + t]);
#else
            dst[(size_t)t * B_DIM * E_DIM + we] = lds[we * SUPER + t];
#endif
        }
        // stores drain via implicit wait-idle at S_ENDPGM
    }
}

extern "C" void kernel_launch(void* const* d_in, const int* in_sizes, int n_in,
                              void* d_out, int out_size, void* d_ws, size_t ws_size,
                              hipStream_t stream) {
    (void)in_sizes; (void)n_in; (void)out_size; (void)ws_size;
    const float* x    = (const float*)d_in[0];
    const float* rawp = (const float*)d_in[1];
    const float* rawn = (const float*)d_in[2];
    float* out = (float*)d_out;

    float2* S     = (float2*)d_ws;                         // 8 MB: [C16][B][E]
    float2* carry = S + (size_t)C16 * B_DIM * E_DIM;       // 8 MB: [B][E][C16]

    k1_summaries<<<(C16 * B_DIM * E_DIM) / 256, 256, 0, stream>>>(x, rawp, rawn, S);
    k2_carries<<<(B_DIM * E_DIM + 255) / 256, 256, 0, stream>>>(rawp, rawn, S, carry);
    k3_wmma_scan<<<(T_DIM / SUPER) * B_DIM * (E_DIM / ETILE), 1024, 0, stream>>>(
        x, rawp, rawn, carry, out);
}